// PoissonNLLwithEMDLoss_43422119363283
// MI455X (gfx1250) — compile-verified
//
#include <hip/hip_runtime.h>

// ---------------------------------------------------------------------------
// PoissonNLL + debiased Sinkhorn-divergence loss, MI455X (gfx1250, wave32).
//
// Phase 1: bandwidth-bound ragged segment reduce (293 MB @ 23.3 TB/s ~ 13us).
// Phase 2: per-(event,OT-kind) standard-domain Sinkhorn with the fixed
//          32x32 Gibbs kernel K = exp(-C/eps) held in VGPRs as WMMA B
//          operands; each scaling mat-vec = 16 x V_WMMA_F32_16X16X4_F32.
//          Reciprocals use hardware v_rcp_f32 (single TRANS op) to keep the
//          serial u->v->u dependency chain short.
// Phase 3: scalar finalize.
// ---------------------------------------------------------------------------

typedef float v2f __attribute__((ext_vector_type(2)));
typedef float v8f __attribute__((ext_vector_type(8)));

#define NPMTS      32
#define EPS_SINK   0.05f
#define INV_EPS    20.0f
#define SINK_ITERS 50
#define POISSON_EPS 1e-8f
#define LOG_FLOOR  1e-30f

__device__ __forceinline__ float wave_sum32(float x) {
#pragma unroll
  for (int o = 16; o > 0; o >>= 1) x += __shfl_xor(x, o, 32);
  return x;
}

// ---------------------------------------------------------------------------
__global__ void zero_ws_kernel(float* p, int n) {
  int i = blockIdx.x * blockDim.x + threadIdx.x;
  if (i < n) p[i] = 0.0f;
}

// ---------------------------------------------------------------------------
// Segment reduce: pe_batch[seg][pmt] += q_mean[v] * pred[v][pmt].
// Block = 256 threads (8 waves). Each wave: 32 iterations, 4 voxels/iter,
// lane l loads float4 of pred (global_load_b128, fully coalesced 512B/wave).
__global__ __launch_bounds__(256) void reduce_kernel(
    const float* __restrict__ pred, const float* __restrict__ charge,
    float* __restrict__ pe_batch, int SEG, int blocks_per_seg) {
  const int seg   = blockIdx.x / blocks_per_seg;
  const int split = blockIdx.x % blocks_per_seg;
  const int w = threadIdx.x >> 5;
  const int l = threadIdx.x & 31;
  const int pm = l & 7;        // pmt quad index: pmts [4*pm, 4*pm+3]
  const int vo = l >> 3;       // voxel offset within quad-group (0..3)

  const long segbase  = (long)seg * SEG;
  const int  wavebase = split * 1024 + w * 128;  // voxel offset in segment

  const float4* __restrict__ pred4 = (const float4*)pred;
  float ax = 0.f, ay = 0.f, az = 0.f, aw = 0.f;

  for (int it = 0; it < 32; ++it) {
    int vrel = wavebase + it * 4 + vo;
    if (vrel < SEG) {
      long vg = segbase + vrel;
      float q = (charge[3 * vg] + charge[3 * vg + 1] + charge[3 * vg + 2]) *
                (1.0f / 3.0f);
      float4 p = pred4[vg * 8 + pm];
      ax += q * p.x; ay += q * p.y; az += q * p.z; aw += q * p.w;
    }
  }

  __shared__ float red[8][128];
  red[w][l * 4 + 0] = ax;
  red[w][l * 4 + 1] = ay;
  red[w][l * 4 + 2] = az;
  red[w][l * 4 + 3] = aw;
  __syncthreads();

  if (threadIdx.x < NPMTS) {
    int p = threadIdx.x;
    float s = 0.f;
#pragma unroll
    for (int ww = 0; ww < 8; ++ww)
#pragma unroll
      for (int g = 0; g < 4; ++g)
        s += red[ww][(g * 8 + (p >> 2)) * 4 + (p & 3)];
    atomicAdd(&pe_batch[seg * NPMTS + p], s);
  }
}

// ---------------------------------------------------------------------------
// One wave per (event, ot_kind). kind 0: OT(a,b) w=+1; 1: OT(a,a) w=-0.5;
// 2: OT(b,a) w=-0.5. Standard-domain Sinkhorn, K in WMMA B-operand layout.
__global__ __launch_bounds__(32) void sinkhorn_kernel(
    const float* __restrict__ pe_batch, const float* __restrict__ target_pe,
    const float* __restrict__ pmtpos, float* __restrict__ accums) {
  const int e    = blockIdx.x / 3;
  const int kind = blockIdx.x % 3;
  const int l    = threadIdx.x;
  const int half = l >> 4;   // which 16-lane half of the wave
  const int n    = l & 15;   // N (column) index within WMMA tile

  // PMT position of this lane's point
  const float pz = pmtpos[2 * l + 0];
  const float py = pmtpos[2 * l + 1];

  // marginals
  const float pe = pe_batch[e * NPMTS + l];
  const float tg = target_pe[e * NPMTS + l];
  const float pes = wave_sum32(pe);
  const float tgs = wave_sum32(tg);
  const float a_full = pe / pes + LOG_FLOOR;
  const float b_full = tg / tgs + LOG_FLOOR;

  // Poisson NLL contribution (once per event, on the kind-0 block)
  if (kind == 0) {
    float term = pe - tg * __logf(pe + POISSON_EPS);
    float tsum = wave_sum32(term);
    if (l == 0) atomicAdd(&accums[1], tsum);
  }

  const float a = (kind == 2) ? b_full : a_full;
  const float b = (kind == 0) ? b_full : a_full;
  const float weight = (kind == 0) ? 1.0f : -0.5f;

  // ---- Precompute K = exp(-C/eps) directly into WMMA B-operand layout.
  // B (4x16 f32): VGPR0 = rows {K=0 | K=2} for lane halves, VGPR1 = {K=1 | K=3}.
  // Chunk c covers K-rows 4c..4c+3; h selects output columns 0-15 / 16-31.
  float Bv0[8][2], Bv1[8][2];
#pragma unroll
  for (int c = 0; c < 8; ++c) {
    const int r0 = 4 * c + 2 * half;
    const float zr0 = __shfl(pz, r0, 32),     yr0 = __shfl(py, r0, 32);
    const float zr1 = __shfl(pz, r0 + 1, 32), yr1 = __shfl(py, r0 + 1, 32);
#pragma unroll
    for (int h = 0; h < 2; ++h) {
      const int col = n + 16 * h;
      const float zc = __shfl(pz, col, 32), yc = __shfl(py, col, 32);
      float d0 = __fsqrt_rn((zr0 - zc) * (zr0 - zc) + (yr0 - yc) * (yr0 - yc) + 1e-12f);
      float d1 = __fsqrt_rn((zr1 - zc) * (zr1 - zc) + (yr1 - yc) * (yr1 - yc) + 1e-12f);
      Bv0[c][h] = __expf(-d0 * INV_EPS);
      Bv1[c][h] = __expf(-d1 * INV_EPS);
    }
  }

  // ---- mat-vec: w[j] = sum_k K[k][j] * t[k]  (K symmetric -> also K t).
  // A rows replicated with t -> every row of D equals the result; lane j of
  // the proper half-tile holds w[j] in D's first VGPR.
  auto matvec = [&](float t) -> float {
    v8f acc0 = {0.f, 0.f, 0.f, 0.f, 0.f, 0.f, 0.f, 0.f};
    v8f acc1 = {0.f, 0.f, 0.f, 0.f, 0.f, 0.f, 0.f, 0.f};
#pragma unroll
    for (int c = 0; c < 8; ++c) {
      float a0 = __shfl(t, 4 * c + 2 * half, 32);
      float a1 = __shfl(t, 4 * c + 1 + 2 * half, 32);
      v2f A  = {a0, a1};
      v2f B0 = {Bv0[c][0], Bv1[c][0]};
      v2f B1 = {Bv0[c][1], Bv1[c][1]};
      acc0 = __builtin_amdgcn_wmma_f32_16x16x4_f32(
          false, A, false, B0, (short)0, acc0, false, false);
      acc1 = __builtin_amdgcn_wmma_f32_16x16x4_f32(
          false, A, false, B1, (short)0, acc1, false, false);
    }
    return (l < 16) ? acc0[0] : acc1[0];
  };

  // ---- Sinkhorn scaling iterations (== log-domain reference updates):
  // u = 1/(K (b.*v)) ; v = 1/(K (a.*u))
  // Hardware v_rcp_f32 (~1 ulp) keeps the serial chain short.
  float u = 1.0f, v = 1.0f;
#pragma unroll 1
  for (int it = 0; it < SINK_ITERS; ++it) {
    u = __builtin_amdgcn_rcpf(matvec(b * v));
    v = __builtin_amdgcn_rcpf(matvec(a * u));
  }

  const float f = EPS_SINK * __logf(u);
  const float g = EPS_SINK * __logf(v);
  const float cost = wave_sum32(a * f + b * g);
  if (l == 0) atomicAdd(&accums[0], weight * cost);
}

// ---------------------------------------------------------------------------
__global__ void finalize_kernel(const float* __restrict__ accums,
                                float* __restrict__ out, float invB,
                                float invBM) {
  out[0] = accums[0] * invB + accums[1] * invBM;
}

// ---------------------------------------------------------------------------
extern "C" void kernel_launch(void* const* d_in, const int* in_sizes, int n_in,
                              void* d_out, int out_size, void* d_ws,
                              size_t ws_size, hipStream_t stream) {
  const float* pred    = (const float*)d_in[0];
  const float* charge  = (const float*)d_in[1];
  const float* target  = (const float*)d_in[2];
  const float* pmtpos  = (const float*)d_in[5];

  const int  B   = in_sizes[3];                 // 64 events
  const long N   = (long)in_sizes[0] / NPMTS;   // 2,097,152 voxels
  const int  SEG = (int)(N / B);                // 32768 voxels/event

  float* ws       = (float*)d_ws;
  float* pe_batch = ws;                      // B*32 floats
  float* accums   = ws + (size_t)B * NPMTS;  // [0]=emd sum, [1]=mag sum

  // zero accumulators every call (graph-replay deterministic)
  const int nz = B * NPMTS + 2;
  zero_ws_kernel<<<(nz + 255) / 256, 256, 0, stream>>>(ws, nz);

  const int vox_per_block  = 1024;  // 8 waves * 128 voxels
  const int blocks_per_seg = (SEG + vox_per_block - 1) / vox_per_block;
  reduce_kernel<<<B * blocks_per_seg, 256, 0, stream>>>(pred, charge, pe_batch,
                                                        SEG, blocks_per_seg);

  sinkhorn_kernel<<<B * 3, 32, 0, stream>>>(pe_batch, target, pmtpos, accums);

  finalize_kernel<<<1, 1, 0, stream>>>(accums, (float*)d_out, 1.0f / (float)B,
                                       1.0f / (float)(B * NPMTS));
}